// SpatialTransformer_17841294148088
// MI455X (gfx1250) — compile-verified
//
#include <hip/hip_runtime.h>
#include <stdint.h>

// Problem constants (from reference): vol [2,2,96,128,128] f32, df [2,3,96,128,128] f32
#define WW   128
#define HH   128
#define DD   96
#define HWC  (HH * WW)          // 16384
#define DHW  (DD * HH * WW)     // 1572864
#define BN   2
#define CN   2
#define NPTS (BN * DHW)         // 3145728
#define TILE 256
#define NTILES (NPTS / TILE)    // 12288  (DHW divisible by TILE -> tiles never straddle batch)
#define TILES_PER_BATCH (DHW / TILE)  // 6144

__global__ __launch_bounds__(256)
void warp3d_kernel(const float* __restrict__ vol,
                   const float* __restrict__ df,
                   float* __restrict__ out)
{
    // Double-buffered df tile: [buf][plane][point] -> 2*3*256*4 = 6 KB LDS
    __shared__ float dtile[2][3][TILE];

    const int tid = threadIdx.x;
    // Generic pointer to __shared__ is {SHARED_BASE[63:32], lds_offset[31:0]};
    // truncation yields the wave-relative LDS byte address the async op needs.
    const uint32_t lds_base = (uint32_t)(uintptr_t)&dtile[0][0][0];
    const uint32_t voff     = (uint32_t)tid * 4u;   // per-lane byte offset within a plane row

    // Issue 3 async global->LDS loads (dx,dy,dz planes) for tile t into buffer `buf`.
    auto issue_tile = [&](int buf, int t) {
        const int b  = t / TILES_PER_BATCH;
        const int s0 = (t % TILES_PER_BATCH) * TILE;
        const uint64_t base = (uint64_t)(uintptr_t)(df + ((size_t)b * 3) * DHW + s0);
#pragma unroll
        for (int p = 0; p < 3; ++p) {
            const uint32_t la = lds_base + (uint32_t)(((buf * 3 + p) * TILE) * 4) + voff;
            const uint64_t sb = base + (uint64_t)p * ((uint64_t)DHW * 4u);
            // GVS mode: mem_addr = SADDR + VADDR[31:0]; VDST = LDS byte address.
            asm volatile("global_load_async_to_lds_b32 %0, %1, %2"
                         :: "v"(la), "v"(voff), "s"(sb)
                         : "memory");
        }
    };

    auto compute_tile = [&](int buf, int t) {
        const int b = t / TILES_PER_BATCH;
        const int s = (t % TILES_PER_BATCH) * TILE + tid;

        const float dx = dtile[buf][0][tid];
        const float dy = dtile[buf][1][tid];
        const float dz = dtile[buf][2][tid];

        const int w = s & (WW - 1);
        const int h = (s >> 7) & (HH - 1);
        const int z = s >> 14;

        // Clamped absolute sample location per dim
        const float fx = fminf(fmaxf((float)w + dx, 0.f), (float)(WW - 1));
        const float fy = fminf(fmaxf((float)h + dy, 0.f), (float)(HH - 1));
        const float fz = fminf(fmaxf((float)z + dz, 0.f), (float)(DD - 1));

        int x0 = (int)floorf(fx); int x1 = x0 + 1; if (x1 > WW - 1) x1 = WW - 1;
        int y0 = (int)floorf(fy); int y1 = y0 + 1; if (y1 > HH - 1) y1 = HH - 1;
        int z0 = (int)floorf(fz); int z1 = z0 + 1; if (z1 > DD - 1) z1 = DD - 1;

        const float tx = fx - (float)x0;  const float wx0 = 1.f - tx;
        const float ty = fy - (float)y0;
        const float tz = fz - (float)z0;

        // Combined z*y weights for the 4 corner rows
        const float wz0 = 1.f - tz;
        const float wy0 = 1.f - ty;
        const float wA = wz0 * wy0;   // (z0,y0)
        const float wB = wz0 * ty;    // (z0,y1)
        const float wC = tz  * wy0;   // (z1,y0)
        const float wD = tz  * ty;    // (z1,y1)

        const int r00 = (z0 << 14) + (y0 << 7);
        const int r01 = (z0 << 14) + (y1 << 7);
        const int r10 = (z1 << 14) + (y0 << 7);
        const int r11 = (z1 << 14) + (y1 << 7);

        const float* __restrict__ v0 = vol + ((size_t)b * CN) * DHW;  // channel 0
        const float* __restrict__ v1 = v0 + DHW;                      // channel 1

        // 16 gathers (vol is fully L2-resident: 25 MB << 192 MB L2)
        const float a000 = v0[r00 + x0], a001 = v0[r00 + x1];
        const float a010 = v0[r01 + x0], a011 = v0[r01 + x1];
        const float a100 = v0[r10 + x0], a101 = v0[r10 + x1];
        const float a110 = v0[r11 + x0], a111 = v0[r11 + x1];
        const float b000 = v1[r00 + x0], b001 = v1[r00 + x1];
        const float b010 = v1[r01 + x0], b011 = v1[r01 + x1];
        const float b100 = v1[r10 + x0], b101 = v1[r10 + x1];
        const float b110 = v1[r11 + x0], b111 = v1[r11 + x1];

        const float acc0 = wA * (wx0 * a000 + tx * a001)
                         + wB * (wx0 * a010 + tx * a011)
                         + wC * (wx0 * a100 + tx * a101)
                         + wD * (wx0 * a110 + tx * a111);
        const float acc1 = wA * (wx0 * b000 + tx * b001)
                         + wB * (wx0 * b010 + tx * b011)
                         + wC * (wx0 * b100 + tx * b101)
                         + wD * (wx0 * b110 + tx * b111);

        out[((size_t)b * CN + 0) * DHW + s] = acc0;
        out[((size_t)b * CN + 1) * DHW + s] = acc1;
    };

    int t = blockIdx.x;
    if (t >= NTILES) return;

    int buf = 0;
    issue_tile(buf, t);                      // prologue: start df DMA for first tile

    for (;;) {
        const int tn = t + gridDim.x;
        if (tn < NTILES) {
            issue_tile(buf ^ 1, tn);         // overlap next tile's df fetch with compute
            // <=3 outstanding: the 3 loads just issued for tn; tile t's loads complete
            // first (async loads return done in order).
            asm volatile("s_wait_asynccnt 3" ::: "memory");
        } else {
            asm volatile("s_wait_asynccnt 0" ::: "memory");
        }
        compute_tile(buf, t);
        if (tn >= NTILES) break;
        t = tn;
        buf ^= 1;
    }
}

extern "C" void kernel_launch(void* const* d_in, const int* in_sizes, int n_in,
                              void* d_out, int out_size, void* d_ws, size_t ws_size,
                              hipStream_t stream)
{
    (void)in_sizes; (void)n_in; (void)out_size; (void)d_ws; (void)ws_size;
    const float* vol = (const float*)d_in[0];
    const float* df  = (const float*)d_in[1];
    float* out       = (float*)d_out;

    const int grid = 1536;   // 8 tiles per block -> double buffering pays off
    warp3d_kernel<<<grid, TILE, 0, stream>>>(vol, df, out);
}